// Net_30039001269040
// MI455X (gfx1250) — compile-verified
//
#include <hip/hip_runtime.h>
#include <hip/hip_bf16.h>

typedef __attribute__((ext_vector_type(2))) float v2f;
typedef __attribute__((ext_vector_type(8))) float v8f;

#define N_NODES 100000
#define N_EDGES 1600000
#define N_FEAT 128
#define HIDDEN 64
#define N_CLASS 40

// ---------------------------------------------------------------------------
// Degree accumulation: deg[dst] += w[e]  (self-loop +1 folded into dinv pass)
// ---------------------------------------------------------------------------
__global__ void deg_kernel(const int* __restrict__ dst, const float* __restrict__ w,
                           float* __restrict__ deg) {
    int e = blockIdx.x * 256 + threadIdx.x;
    if (e < N_EDGES) atomicAdd(&deg[dst[e]], w[e]);
}

__global__ void dinv_kernel(float* __restrict__ deg) {
    int i = blockIdx.x * 256 + threadIdx.x;
    if (i < N_NODES) {
        float d = deg[i] + 1.0f;           // + self-loop weight
        deg[i] = rsqrtf(d);                // deg>=1 > 0 always
    }
}

// ---------------------------------------------------------------------------
// GEMM1: H1[100000x64] = X[100000x128] @ W1[128x64], fp32 WMMA 16x16x4
// Block = 256 thr = 8 waves -> 32x64 output region (2x4 tiles of 16x16).
// A-frag (16x4 f32): lane<16 holds row M=lane, K = k0..k0+1; lane>=16 K=k0+2..3
// B/C/D: row striped across lanes within a VGPR (ISA 7.12.2).
// ---------------------------------------------------------------------------
__global__ __launch_bounds__(256) void gemm1_kernel(const float* __restrict__ X,
                                                    const float* __restrict__ W,
                                                    float* __restrict__ H) {
    __shared__ float sW[128 * 65];   // stride 65 to avoid bank conflicts
    __shared__ float sA[32 * 130];   // stride 130 (even: keeps b64 alignment)
    const int tid = threadIdx.x;
    const int rowbase = blockIdx.x * 32;

    for (int idx = tid; idx < 128 * 64; idx += 256) {
        int r = idx >> 6, c = idx & 63;
        sW[r * 65 + c] = W[idx];
    }
    for (int idx = tid; idx < 32 * 128; idx += 256) {
        int r = idx >> 7, c = idx & 127;
        sA[r * 130 + c] = X[(size_t)(rowbase + r) * N_FEAT + c];
    }
    __syncthreads();

    const int wv = tid >> 5, lane = tid & 31;
    const int tr = wv >> 2, tc = wv & 3;           // tile row (0..1), tile col (0..3)
    const int mm = tr * 16 + (lane & 15);          // A row for this lane
    const int nn = tc * 16 + (lane & 15);          // B col for this lane
    const int kh = (lane >> 4) * 2;                // K sub-offset per half-wave

    v8f acc = {};
    for (int k0 = 0; k0 < 128; k0 += 4) {
        v2f a, b;
        a.x = sA[mm * 130 + k0 + kh];
        a.y = sA[mm * 130 + k0 + kh + 1];
        b.x = sW[(k0 + kh) * 65 + nn];
        b.y = sW[(k0 + kh + 1) * 65 + nn];
        acc = __builtin_amdgcn_wmma_f32_16x16x4_f32(false, a, false, b,
                                                    (short)0, acc, false, false);
    }
    // C/D layout: VGPR r -> M=r (lanes 0-15) / M=r+8 (lanes 16-31), N=lane&15
    const int half = (lane >> 4) * 8;
    const int col  = tc * 16 + (lane & 15);
    for (int r = 0; r < 8; ++r) {
        int row = rowbase + tr * 16 + half + r;
        H[(size_t)row * HIDDEN + col] = acc[r];
    }
}

// ---------------------------------------------------------------------------
// Edge scatter layer 1: agg1[dst, f] += dinv[src]*w*dinv[dst] * h1[src, f]
// 64-thread group per edge -> coalesced 256B gather of h1[src], L2 atomics out.
// ---------------------------------------------------------------------------
__global__ void scatter1_kernel(const int* __restrict__ src, const int* __restrict__ dst,
                                const float* __restrict__ w, const float* __restrict__ dinv,
                                const float* __restrict__ h1, float* __restrict__ agg1) {
    long long idx = (long long)blockIdx.x * 256 + threadIdx.x;
    int e = (int)(idx >> 6);
    int f = (int)(idx & 63);
    int s = src[e], d = dst[e];
    float norm = dinv[s] * w[e] * dinv[d];
    atomicAdd(&agg1[(size_t)d * HIDDEN + f], norm * h1[(size_t)s * HIDDEN + f]);
}

// Self-loop (norm = dinv^2) + bias + ReLU, in place on agg1 (GEMM2 input).
__global__ void node1_kernel(float* __restrict__ agg1, const float* __restrict__ h1,
                             const float* __restrict__ dinv, const float* __restrict__ b1) {
    size_t idx = (size_t)blockIdx.x * 256 + threadIdx.x;
    int i = (int)(idx >> 6);
    int f = (int)(idx & 63);
    float di = dinv[i];
    float v = agg1[idx] + di * di * h1[idx] + b1[f];
    agg1[idx] = fmaxf(v, 0.0f);
}

// ---------------------------------------------------------------------------
// GEMM2: H2[100000x40] = agg1[100000x64] @ W2[64x40], N padded to 48 (3 tiles).
// 6 compute waves (2x3 tiles of 16x16); guard is wave-uniform so EXEC==all-1s.
// ---------------------------------------------------------------------------
__global__ __launch_bounds__(256) void gemm2_kernel(const float* __restrict__ A,
                                                    const float* __restrict__ W,
                                                    float* __restrict__ H) {
    __shared__ float sW[64 * 49];    // 64 x 48 padded, stride 49
    __shared__ float sA[32 * 66];    // 32 x 64, stride 66
    const int tid = threadIdx.x;
    const int rowbase = blockIdx.x * 32;

    for (int idx = tid; idx < 64 * 48; idx += 256) {
        int r = idx / 48, c = idx % 48;
        sW[r * 49 + c] = (c < N_CLASS) ? W[r * N_CLASS + c] : 0.0f;
    }
    for (int idx = tid; idx < 32 * 64; idx += 256) {
        int r = idx >> 6, c = idx & 63;
        sA[r * 66 + c] = A[(size_t)(rowbase + r) * HIDDEN + c];
    }
    __syncthreads();

    const int wv = tid >> 5, lane = tid & 31;
    if (wv < 6) {                                  // wave-uniform branch
        const int tr = wv / 3, tc = wv % 3;
        const int mm = tr * 16 + (lane & 15);
        const int nn = tc * 16 + (lane & 15);
        const int kh = (lane >> 4) * 2;
        v8f acc = {};
        for (int k0 = 0; k0 < 64; k0 += 4) {
            v2f a, b;
            a.x = sA[mm * 66 + k0 + kh];
            a.y = sA[mm * 66 + k0 + kh + 1];
            b.x = sW[(k0 + kh) * 49 + nn];
            b.y = sW[(k0 + kh + 1) * 49 + nn];
            acc = __builtin_amdgcn_wmma_f32_16x16x4_f32(false, a, false, b,
                                                        (short)0, acc, false, false);
        }
        const int half = (lane >> 4) * 8;
        const int col  = tc * 16 + (lane & 15);
        if (col < N_CLASS) {
            for (int r = 0; r < 8; ++r) {
                int row = rowbase + tr * 16 + half + r;
                H[(size_t)row * N_CLASS + col] = acc[r];
            }
        }
    }
}

// ---------------------------------------------------------------------------
// Edge scatter layer 2: out[dst, c] += norm * h2[src, c]  (40 thr per edge)
// blockDim = 320 (10 wave32) -> exactly 8 edges per block.
// ---------------------------------------------------------------------------
__global__ void scatter2_kernel(const int* __restrict__ src, const int* __restrict__ dst,
                                const float* __restrict__ w, const float* __restrict__ dinv,
                                const float* __restrict__ h2, float* __restrict__ out) {
    int e = blockIdx.x * 8 + threadIdx.x / N_CLASS;
    int c = threadIdx.x % N_CLASS;
    int s = src[e], d = dst[e];
    float norm = dinv[s] * w[e] * dinv[d];
    atomicAdd(&out[(size_t)d * N_CLASS + c], norm * h2[(size_t)s * N_CLASS + c]);
}

// ---------------------------------------------------------------------------
// Final: self-loop + bias + log_softmax, one wave32 per node (40 classes:
// lane covers c=lane, and c=lane+32 for lane<8). Butterfly shfl reductions.
// ---------------------------------------------------------------------------
__global__ __launch_bounds__(256) void final_kernel(float* __restrict__ out,
                                                    const float* __restrict__ h2,
                                                    const float* __restrict__ dinv,
                                                    const float* __restrict__ b2) {
    const int lane = threadIdx.x & 31;
    const int node = blockIdx.x * 8 + (threadIdx.x >> 5);
    const float di = dinv[node];
    const float dd = di * di;
    const size_t base = (size_t)node * N_CLASS;

    float v0 = out[base + lane] + dd * h2[base + lane] + b2[lane];
    float v1 = -__builtin_huge_valf();
    if (lane < 8)
        v1 = out[base + 32 + lane] + dd * h2[base + 32 + lane] + b2[32 + lane];

    float m = fmaxf(v0, v1);
    for (int off = 16; off; off >>= 1) m = fmaxf(m, __shfl_xor(m, off, 32));
    float s = __expf(v0 - m) + ((lane < 8) ? __expf(v1 - m) : 0.0f);
    for (int off = 16; off; off >>= 1) s += __shfl_xor(s, off, 32);
    float lse = __logf(s);

    out[base + lane] = v0 - m - lse;
    if (lane < 8) out[base + 32 + lane] = v1 - m - lse;
}

// ---------------------------------------------------------------------------
extern "C" void kernel_launch(void* const* d_in, const int* in_sizes, int n_in,
                              void* d_out, int out_size, void* d_ws, size_t ws_size,
                              hipStream_t stream) {
    const float* x   = (const float*)d_in[0];
    const int*   ei  = (const int*)d_in[1];       // [2, E] flat: src then dst
    const float* ew  = (const float*)d_in[2];
    const float* W1  = (const float*)d_in[3];
    const float* b1  = (const float*)d_in[4];
    const float* W2  = (const float*)d_in[5];
    const float* b2  = (const float*)d_in[6];
    float* out = (float*)d_out;

    const int* src = ei;
    const int* dst = ei + N_EDGES;

    // Workspace layout (floats)
    float* ws   = (float*)d_ws;
    float* dinv = ws;                               // 100000 (deg -> dinv in place)
    float* h1   = ws + (1 << 17);                   // 6,400,000
    float* agg1 = h1 + (size_t)N_NODES * HIDDEN;    // 6,400,000
    float* h2   = agg1 + (size_t)N_NODES * HIDDEN;  // 4,000,000

    // Zero accumulators (capture-safe async memsets)
    hipMemsetAsync(dinv, 0, (size_t)N_NODES * sizeof(float), stream);
    hipMemsetAsync(agg1, 0, (size_t)N_NODES * HIDDEN * sizeof(float), stream);
    hipMemsetAsync(out,  0, (size_t)N_NODES * N_CLASS * sizeof(float), stream);

    // Degrees + dinv
    deg_kernel<<<(N_EDGES + 255) / 256, 256, 0, stream>>>(dst, ew, dinv);
    dinv_kernel<<<(N_NODES + 255) / 256, 256, 0, stream>>>(dinv);

    // Layer 1
    gemm1_kernel<<<N_NODES / 32, 256, 0, stream>>>(x, W1, h1);          // 3125 blocks
    scatter1_kernel<<<(N_EDGES * 64) / 256, 256, 0, stream>>>(src, dst, ew, dinv, h1, agg1);
    node1_kernel<<<(N_NODES * HIDDEN) / 256, 256, 0, stream>>>(agg1, h1, dinv, b1);

    // Layer 2
    gemm2_kernel<<<N_NODES / 32, 256, 0, stream>>>(agg1, W2, h2);
    scatter2_kernel<<<N_EDGES / 8, 320, 0, stream>>>(src, dst, ew, dinv, h2, out);

    // Bias + self-loop + log_softmax
    final_kernel<<<N_NODES / 8, 256, 0, stream>>>(out, h2, dinv, b2);
}